// CenterLoss_18648747999646
// MI455X (gfx1250) — compile-verified
//
#include <hip/hip_runtime.h>
#include <stdint.h>

// ---------------- problem constants (match reference) ----------------
#define NUM_CLASSES 1000
#define FEAT_DIM    512
#define BATCH       32768
#define ALPHA       0.5f

// ---------------- tiling ----------------
#define CHUNK   2048                 // labels staged per chunk (8 KB)
#define NCHUNK  (BATCH / CHUNK)      // 16 chunks
#define BLOCK   128                  // 4 wave32 waves; each wave owns 128 dims
#define ISSUES  (CHUNK / 128)        // async b128 issues per chunk (wave0: 32 lanes x 16B x 16 = 8 KB)

typedef float v4f __attribute__((ext_vector_type(4)));

// CDNA5 async copy: global memory -> LDS via the async engine (tracked by ASYNCcnt).
// vdst VGPR = per-lane LDS byte offset, vaddr = per-lane 64-bit global address.
__device__ __forceinline__ void async_lds_b128(uint32_t lds_off, const void* gaddr) {
  asm volatile("global_load_async_to_lds_b128 %0, %1, off"
               :: "v"(lds_off), "v"(gaddr)
               : "memory");
}
__device__ __forceinline__ void wait_async_le0()  { asm volatile("s_wait_asynccnt 0x0"  ::: "memory"); }
__device__ __forceinline__ void wait_async_le16() { asm volatile("s_wait_asynccnt 0x10" ::: "memory"); }

// One block per class c:
//   * stream all labels through LDS (async, double-buffered), ballot-scan for label==c
//   * gather matching feature rows (nontemporal b128), accumulate per-lane f32 sums + sum-of-squares
//   * write new_centers[c] directly; contribute one scalar to the loss via the algebraic identity
//       sum_b ||f_b - cen_{l_b}||^2 = sum_c [ S2_c - 2*sums_c.cen_c + count_c*||cen_c||^2 ]
__global__ void __launch_bounds__(BLOCK)
center_loss_fused(const float* __restrict__ features,
                  const int*   __restrict__ labels,
                  const float* __restrict__ centers,
                  float*       __restrict__ out) {
  __shared__ int   sh[2][CHUNK];
  __shared__ float red[BLOCK];

  const int c    = blockIdx.x;              // this block's class
  const int tid  = threadIdx.x;
  const int lane = tid & 31;
  const int dim  = ((tid >> 5) << 7) + (lane << 2);   // wave*128 + lane*4 : this thread's 4 dims

  // LDS byte offset of the staging buffer (generic LDS pointer low 32 bits = LDS offset)
  const uint32_t sh_base = (uint32_t)(uintptr_t)(&sh[0][0]);

  // ---- preload chunk 0 (wave0 issues; 32 lanes x 16B per instruction) ----
  if (tid < 32) {
    for (int i = 0; i < ISSUES; ++i)
      async_lds_b128(sh_base + (uint32_t)(i * 512 + lane * 16),
                     labels + i * 128 + lane * 4);
  }

  v4f   acc = {0.f, 0.f, 0.f, 0.f};   // per-lane slice of sums[c]
  float sq  = 0.f;                    // per-lane slice of sum_{b in c} f^2
  int   cnt = 0;                      // count_c (wave-uniform)

  for (int k = 0; k < NCHUNK; ++k) {
    if (tid < 32) {
      if (k + 1 < NCHUNK) {
        // prefetch next chunk into the other buffer, then wait for chunk k only.
        // Async loads complete in order, so ASYNCcnt<=16 (the 16 just-issued ops)
        // guarantees all 16 ops of chunk k have landed in LDS.
        const uint32_t bb = sh_base + (uint32_t)(((k + 1) & 1) * (CHUNK * 4));
        const int* gs = labels + (k + 1) * CHUNK;
        for (int i = 0; i < ISSUES; ++i)
          async_lds_b128(bb + (uint32_t)(i * 512 + lane * 16), gs + i * 128 + lane * 4);
        wait_async_le16();
      } else {
        wait_async_le0();
      }
    }
    __syncthreads();   // chunk k visible to all 4 waves

    const int* lab  = &sh[k & 1][0];
    const int  base = k * CHUNK;
    for (int r = 0; r < CHUNK / 32; ++r) {
      const int l = lab[(r << 5) + lane];
      unsigned mask = __builtin_amdgcn_ballot_w32(l == c);   // wave-uniform SGPR mask
      cnt += __builtin_popcount(mask);
      while (mask) {                                         // uniform loop: usually 0-1 iters
        const int b = base + (r << 5) + __builtin_ctz(mask);
        mask &= mask - 1;
        // each feature row is touched by exactly one block -> nontemporal, keep L2 for labels/centers
        const v4f f = __builtin_nontemporal_load(
            (const v4f*)(features + (size_t)b * FEAT_DIM + dim));
        acc += f;
        sq  += f.x * f.x + f.y * f.y + f.z * f.z + f.w * f.w;
      }
    }
    __syncthreads();   // all waves done with buf[k&1] before it is overwritten at iter k+1
  }

  // ---- loss partial + center update for class c ----
  const v4f c4 = *(const v4f*)(centers + (size_t)c * FEAT_DIM + dim);
  const float fcnt = (float)cnt;
  float part = sq
             - 2.0f * (acc.x * c4.x + acc.y * c4.y + acc.z * c4.z + acc.w * c4.w)
             + fcnt * (c4.x * c4.x + c4.y * c4.y + c4.z * c4.z + c4.w * c4.w);
  red[tid] = part;
  __syncthreads();
  #pragma unroll
  for (int s = BLOCK / 2; s > 0; s >>= 1) {
    if (tid < s) red[tid] += red[tid + s];
    __syncthreads();
  }
  if (tid == 0) atomicAdd(out, red[0] * (0.5f / (float)BATCH));   // 1000 atomics total

  // out layout: [0] = loss, [1 .. 1+C*D) = new_centers (unaligned by 4B -> scalar stores)
  float* oc = out + 1 + (size_t)c * FEAT_DIM + dim;
  if (cnt > 0) {
    const float inv = 1.0f / fcnt;
    oc[0] = (1.0f - ALPHA) * c4.x + ALPHA * (acc.x * inv);
    oc[1] = (1.0f - ALPHA) * c4.y + ALPHA * (acc.y * inv);
    oc[2] = (1.0f - ALPHA) * c4.z + ALPHA * (acc.z * inv);
    oc[3] = (1.0f - ALPHA) * c4.w + ALPHA * (acc.w * inv);
  } else {
    oc[0] = c4.x; oc[1] = c4.y; oc[2] = c4.z; oc[3] = c4.w;
  }
}

__global__ void zero_loss(float* out) { if (threadIdx.x == 0) out[0] = 0.0f; }

extern "C" void kernel_launch(void* const* d_in, const int* in_sizes, int n_in,
                              void* d_out, int out_size, void* d_ws, size_t ws_size,
                              hipStream_t stream) {
  const float* features = (const float*)d_in[0];   // [32768, 512] f32
  const int*   labels   = (const int*)d_in[1];     // [32768] int
  const float* centers  = (const float*)d_in[2];   // [1000, 512] f32
  float* out = (float*)d_out;                      // [1 + 1000*512] f32

  zero_loss<<<1, 32, 0, stream>>>(out);
  center_loss_fused<<<NUM_CLASSES, BLOCK, 0, stream>>>(features, labels, centers, out);

  (void)in_sizes; (void)n_in; (void)out_size; (void)d_ws; (void)ws_size;
}